// SSM_73907797229639
// MI455X (gfx1250) — compile-verified
//
#include <hip/hip_runtime.h>
#include <math.h>

// ---------------------------------------------------------------------------
// Problem constants (all dims 256)
// ---------------------------------------------------------------------------
#define TSTEPS 256
#define BATCH  256
#define DIM    256                       // NX == NY == NU == ND
#define TB     (TSTEPS * BATCH)          // 65536 rows
#define PLANE  (TB * DIM)                // 16777216 elements of a [T,B,256] tensor

#define HEAT_C   1.1591509722222224f
#define XMAXC    35.0f
#define UMINC    (-5000.0f)
#define UMAXC    5000.0f
#define DXMINC   (-0.5f)
#define DXMAXC   0.5f
#define Q_DX     (100.0f    / 128.0f)
#define Q_DX_UD  (100000.0f / 128.0f)
#define Q_CON_X  (10.0f     / 128.0f)
#define Q_CON_U  (10.0f     / 256.0f)

// d_out layout (floats): X | Y | U | reg
#define X_OFF   0
#define Y_OFF   PLANE
#define U_OFF   (2 * PLANE)
#define REG_OFF (3 * PLANE)

// ws layout (floats)
#define WS_X0        0                       // B*DIM = 65536
#define N_UBLOCKS    16384
#define WS_UPART     (WS_X0 + BATCH * DIM)   // N_UBLOCKS * 2
#define N_GBLOCKS    2048
#define WS_INPPART   (WS_UPART + N_UBLOCKS * 2)
#define N_SBLOCKS    64                      // per time step
#define WS_SCANPART  (WS_INPPART + N_GBLOCKS * 2)  // TSTEPS * N_SBLOCKS * 3

typedef __attribute__((ext_vector_type(2))) float v2f;
typedef __attribute__((ext_vector_type(8))) float v8f;

__device__ __forceinline__ v8f zero8() {
  v8f z;
#pragma unroll
  for (int i = 0; i < 8; ++i) z[i] = 0.0f;
  return z;
}

// D = A(16x4) * B(4x16) + C ; f32, preserves reference fp32 semantics.
__device__ __forceinline__ v8f wmma4(v2f a, v2f b, v8f c) {
  return __builtin_amdgcn_wmma_f32_16x16x4_f32(false, a, false, b, (short)0, c,
                                               false, false);
}

// A- and B-fragment loader for V_WMMA_F32_16X16X4_F32 (ISA 7.12.2):
// lane L holds (row = rowBase + L%16, k = kBase + 2*(L>>4) + r) in vgpr r.
// Both operands of out[m,n] = sum_k In[m,k]*W[n,k] use this mapping
// (row==m for A from In, row==n for B from W, since W is stored [n,k]).
__device__ __forceinline__ v2f frag_ld(const float* __restrict__ base, int ld,
                                       int row, int k) {
  const int lane = threadIdx.x & 31;
  const float* p = base + (row + (lane & 15)) * ld + (k + ((lane >> 4) << 1));
  v2f r;
  r[0] = p[0];
  r[1] = p[1];
  return r;
}

__device__ __forceinline__ void breduce(float v, float* sm, float* dst, int nt) {
  sm[threadIdx.x] = v;
  __syncthreads();
  for (int s = nt >> 1; s > 0; s >>= 1) {
    if ((int)threadIdx.x < s) sm[threadIdx.x] += sm[threadIdx.x + s];
    __syncthreads();
  }
  if (threadIdx.x == 0) *dst = sm[0];
  __syncthreads();
}

// ---------------------------------------------------------------------------
// 0) x0 = x + x0_correct (broadcast over batch)
// ---------------------------------------------------------------------------
__global__ __launch_bounds__(256) void k_x0(const float* __restrict__ x,
                                            const float* __restrict__ xc,
                                            float* __restrict__ x0) {
  const int i = blockIdx.x * 256 + threadIdx.x;  // 65536 total
  x0[i] = x[i] + xc[i & (DIM - 1)];
}

// ---------------------------------------------------------------------------
// 1) U = HEAT_C * M_flow * DT  (+ fused sumin/sumax penalty partials)
// ---------------------------------------------------------------------------
__global__ __launch_bounds__(256) void k_u(const float* __restrict__ mf,
                                           const float* __restrict__ dt,
                                           float* __restrict__ uout,
                                           float* __restrict__ part) {
  __shared__ float sm[256];
  const int base = (blockIdx.x * 256 + threadIdx.x) * 4;
  const float4 m = *(const float4*)(mf + base);
  const float4 d = *(const float4*)(dt + base);
  float4 u;
  u.x = HEAT_C * m.x * d.x;
  u.y = HEAT_C * m.y * d.y;
  u.z = HEAT_C * m.z * d.z;
  u.w = HEAT_C * m.w * d.w;
  *(float4*)(uout + base) = u;

  float smin = 0.0f, smax = 0.0f;
  const float uv[4] = {u.x, u.y, u.z, u.w};
#pragma unroll
  for (int i = 0; i < 4; ++i) {
    const float t1 = fmaxf(UMINC - uv[i], 0.0f) - UMINC;  // (sumin - UMIN)
    const float t2 = fmaxf(uv[i] - UMAXC, 0.0f) - UMAXC;  // (sumax - UMAX)
    smin += t1 * t1;
    smax += t2 * t2;
  }
  breduce(smin, sm, part + blockIdx.x * 2 + 0, 256);
  breduce(smax, sm, part + blockIdx.x * 2 + 1, 256);
}

// ---------------------------------------------------------------------------
// 2) inp = U @ Bw^T + D @ Ew^T  (WMMA f32), fused dx_u/dx_d penalties.
//    Block: 256 thr = 8 waves; block tile 128(M) x 64(N); wave tile 16 x 64.
// ---------------------------------------------------------------------------
__global__ __launch_bounds__(256) void k_inp(const float* __restrict__ U,
                                             const float* __restrict__ D,
                                             const float* __restrict__ Bw,
                                             const float* __restrict__ Ew,
                                             float* __restrict__ inp,
                                             float* __restrict__ part) {
  __shared__ float sm[256];
  const int col = (blockIdx.x & 3) * 64;
  const int row = (blockIdx.x >> 2) * 128 + (threadIdx.x >> 5) * 16;

  v8f accU[4], accD[4];
#pragma unroll
  for (int j = 0; j < 4; ++j) { accU[j] = zero8(); accD[j] = zero8(); }

  for (int k = 0; k < DIM; k += 4) {
    const v2f aU = frag_ld(U, DIM, row, k);
    const v2f aD = frag_ld(D, DIM, row, k);
#pragma unroll
    for (int j = 0; j < 4; ++j) {
      const v2f bB = frag_ld(Bw, DIM, col + j * 16, k);
      const v2f bE = frag_ld(Ew, DIM, col + j * 16, k);
      accU[j] = wmma4(aU, bB, accU[j]);
      accD[j] = wmma4(aD, bE, accD[j]);
    }
  }

  const int lane = threadIdx.x & 31;
  const int n = lane & 15;
  const int mo = (lane >> 4) << 3;
  float sdu = 0.0f, sdd = 0.0f;
#pragma unroll
  for (int j = 0; j < 4; ++j) {
#pragma unroll
    for (int r = 0; r < 8; ++r) {
      const float bu = accU[j][r];
      const float ed = accD[j][r];
      const float du = fmaxf(DXMINC - bu, 0.0f) + fmaxf(bu - DXMAXC, 0.0f);
      const float dd = fmaxf(DXMINC - ed, 0.0f) + fmaxf(ed - DXMAXC, 0.0f);
      sdu += du * du;
      sdd += dd * dd;
      inp[(row + mo + r) * DIM + col + j * 16 + n] = bu + ed;
    }
  }
  breduce(sdu, sm, part + blockIdx.x * 2 + 0, 256);
  breduce(sdd, sm, part + blockIdx.x * 2 + 1, 256);
}

// ---------------------------------------------------------------------------
// 3) One scan step: X[t] = Xprev @ Aw^T + inp[t]  (+ sxmin/sxmax/sdx partials)
//    64 blocks x 128 thr (4 waves); one 16x16 tile per wave, K = 256.
// ---------------------------------------------------------------------------
__global__ __launch_bounds__(128) void k_scan(const float* __restrict__ xprev,
                                              const float* __restrict__ Aw,
                                              const float* __restrict__ inp_t,
                                              float* __restrict__ xout,
                                              float* __restrict__ part) {
  __shared__ float sm[128];
  const int tile = blockIdx.x * 4 + (threadIdx.x >> 5);  // 0..255
  const int tm = (tile >> 4) << 4;
  const int tn = (tile & 15) << 4;

  v8f acc = zero8();
  for (int k = 0; k < DIM; k += 4) {
    const v2f a = frag_ld(xprev, DIM, tm, k);
    const v2f b = frag_ld(Aw, DIM, tn, k);
    acc = wmma4(a, b, acc);
  }

  const int lane = threadIdx.x & 31;
  const int n = lane & 15;
  const int mo = (lane >> 4) << 3;
  float sxm = 0.0f, sxM = 0.0f, sdx = 0.0f;
#pragma unroll
  for (int r = 0; r < 8; ++r) {
    const int idx = (tm + mo + r) * DIM + tn + n;
    const float xv = acc[r] + inp_t[idx];
    const float pv = xprev[idx];
    const float a1 = fmaxf(-xv, 0.0f);            // (sxmin - XMIN), XMIN = 0
    const float a2 = fmaxf(xv - XMAXC, 0.0f) - XMAXC;
    const float dl = xv - pv;
    sxm += a1 * a1;
    sxM += a2 * a2;
    sdx += dl * dl;
    xout[idx] = xv;
  }
  breduce(sxm, sm, part + blockIdx.x * 3 + 0, 128);
  breduce(sxM, sm, part + blockIdx.x * 3 + 1, 128);
  breduce(sdx, sm, part + blockIdx.x * 3 + 2, 128);
}

// ---------------------------------------------------------------------------
// 4) Y = X @ Cw^T  (WMMA f32); overwrites the now-dead inp region.
// ---------------------------------------------------------------------------
__global__ __launch_bounds__(256) void k_y(const float* __restrict__ X,
                                           const float* __restrict__ Cw,
                                           float* __restrict__ Y) {
  const int col = (blockIdx.x & 3) * 64;
  const int row = (blockIdx.x >> 2) * 128 + (threadIdx.x >> 5) * 16;

  v8f acc[4];
#pragma unroll
  for (int j = 0; j < 4; ++j) acc[j] = zero8();

  for (int k = 0; k < DIM; k += 4) {
    const v2f a = frag_ld(X, DIM, row, k);
#pragma unroll
    for (int j = 0; j < 4; ++j) {
      const v2f b = frag_ld(Cw, DIM, col + j * 16, k);
      acc[j] = wmma4(a, b, acc[j]);
    }
  }

  const int lane = threadIdx.x & 31;
  const int n = lane & 15;
  const int mo = (lane >> 4) << 3;
#pragma unroll
  for (int j = 0; j < 4; ++j)
#pragma unroll
    for (int r = 0; r < 8; ++r)
      Y[(row + mo + r) * DIM + col + j * 16 + n] = acc[j][r];
}

// ---------------------------------------------------------------------------
// 5) Deterministic final reduction of all partials -> reg scalar.
// ---------------------------------------------------------------------------
__global__ __launch_bounds__(256) void k_reduce(const float* __restrict__ ws,
                                                float* __restrict__ out) {
  __shared__ float red[256];
  float s[7];
#pragma unroll
  for (int v = 0; v < 7; ++v) s[v] = 0.0f;
  // order: 0 sxmin, 1 sxmax, 2 sumin, 3 sumax, 4 sdx, 5 dx_u, 6 dx_d
  for (int i = threadIdx.x; i < N_UBLOCKS; i += 256) {
    s[2] += ws[WS_UPART + i * 2 + 0];
    s[3] += ws[WS_UPART + i * 2 + 1];
  }
  for (int i = threadIdx.x; i < N_GBLOCKS; i += 256) {
    s[5] += ws[WS_INPPART + i * 2 + 0];
    s[6] += ws[WS_INPPART + i * 2 + 1];
  }
  for (int i = threadIdx.x; i < TSTEPS * N_SBLOCKS; i += 256) {
    s[0] += ws[WS_SCANPART + i * 3 + 0];
    s[1] += ws[WS_SCANPART + i * 3 + 1];
    s[4] += ws[WS_SCANPART + i * 3 + 2];
  }
  float tot[7];
#pragma unroll
  for (int v = 0; v < 7; ++v) {
    red[threadIdx.x] = s[v];
    __syncthreads();
    for (int st = 128; st > 0; st >>= 1) {
      if ((int)threadIdx.x < st) red[threadIdx.x] += red[threadIdx.x + st];
      __syncthreads();
    }
    tot[v] = red[0];
    __syncthreads();
  }
  if (threadIdx.x == 0) {
    const float inv = 1.0f / (float)PLANE;  // every mse is over T*B*256 elems
    out[0] = Q_CON_X * tot[0] * inv + Q_CON_X * tot[1] * inv +
             Q_CON_U * tot[2] * inv + Q_CON_U * tot[3] * inv +
             Q_DX * tot[4] * inv + Q_DX_UD * tot[5] * inv +
             Q_DX_UD * tot[6] * inv;
  }
}

// ---------------------------------------------------------------------------
extern "C" void kernel_launch(void* const* d_in, const int* in_sizes, int n_in,
                              void* d_out, int out_size, void* d_ws,
                              size_t ws_size, hipStream_t stream) {
  (void)in_sizes; (void)n_in; (void)out_size; (void)ws_size;
  const float* x  = (const float*)d_in[0];
  const float* mf = (const float*)d_in[1];
  const float* dt = (const float*)d_in[2];
  const float* D  = (const float*)d_in[3];
  const float* Aw = (const float*)d_in[4];
  const float* Bw = (const float*)d_in[5];
  const float* Ew = (const float*)d_in[6];
  const float* Cw = (const float*)d_in[7];
  const float* xc = (const float*)d_in[8];
  float* out = (float*)d_out;
  float* ws  = (float*)d_ws;

  k_x0<<<BATCH, 256, 0, stream>>>(x, xc, ws + WS_X0);
  k_u<<<N_UBLOCKS, 256, 0, stream>>>(mf, dt, out + U_OFF, ws + WS_UPART);
  // inp lives temporarily in the (not yet written) Y region of d_out.
  k_inp<<<N_GBLOCKS, 256, 0, stream>>>(out + U_OFF, D, Bw, Ew, out + Y_OFF,
                                       ws + WS_INPPART);
  for (int t = 0; t < TSTEPS; ++t) {
    const float* xprev =
        (t == 0) ? (ws + WS_X0) : (out + X_OFF + (size_t)(t - 1) * BATCH * DIM);
    k_scan<<<N_SBLOCKS, 128, 0, stream>>>(
        xprev, Aw, out + Y_OFF + (size_t)t * BATCH * DIM,
        out + X_OFF + (size_t)t * BATCH * DIM,
        ws + WS_SCANPART + (size_t)t * N_SBLOCKS * 3);
  }
  k_y<<<N_GBLOCKS, 256, 0, stream>>>(out + X_OFF, Cw, out + Y_OFF);
  k_reduce<<<1, 256, 0, stream>>>(ws, out + REG_OFF);
}